// BasicCountNet_6871947674185
// MI455X (gfx1250) — compile-verified
//
#include <hip/hip_runtime.h>

typedef float v2f __attribute__((ext_vector_type(2)));
typedef float v8f __attribute__((ext_vector_type(8)));

// ---------------------------------------------------------------------------
// Utility kernels
// ---------------------------------------------------------------------------
__global__ void zero_f(float* __restrict__ p, int n) {
  int i = blockIdx.x * blockDim.x + threadIdx.x;
  if (i < n) p[i] = 0.0f;
}

__global__ void copy_f4(const float* __restrict__ src, float* __restrict__ dst, int n4) {
  const float4* s = (const float4*)src;
  float4* d = (float4*)dst;
  int stride = gridDim.x * blockDim.x;
  for (int i = blockIdx.x * blockDim.x + threadIdx.x; i < n4; i += stride)
    d[i] = s[i];
}

// WT[c*K + k] = W[k*M + c]; K, M multiples of 16. 256-thread blocks, 16x16 tiles.
__global__ void transpose_w(const float* __restrict__ W, float* __restrict__ WT,
                            int K, int M) {
  __shared__ float tile[16][17];
  const int kb = blockIdx.x * 16;
  const int cb = blockIdx.y * 16;
  const int tx = threadIdx.x & 15;
  const int ty = threadIdx.x >> 4;
  tile[ty][tx] = W[(unsigned)(kb + ty) * M + (cb + tx)];
  __syncthreads();
  WT[(unsigned)(cb + ty) * K + (kb + tx)] = tile[tx][ty];
}

// z[dst] += x[src] over all edges; feature width F = 4 << gshift floats.
// edges layout: edges[0..E-1] = src, edges[E..2E-1] = dst.
__global__ void scatter_add_edges(const float* __restrict__ x,
                                  const int* __restrict__ edges,
                                  int E, int gshift, float* __restrict__ z) {
  const int gmask = (1 << gshift) - 1;
  const int total = E << gshift;
  const int stride = gridDim.x * blockDim.x;
  const int fshift = gshift + 2;             // log2(floats per row)
  for (int i = blockIdx.x * blockDim.x + threadIdx.x; i < total; i += stride) {
    const int e = i >> gshift;
    const int g = i & gmask;
    const unsigned src = (unsigned)edges[e];
    const unsigned dst = (unsigned)edges[E + e];
    const float4 v = *(const float4*)(x + (src << fshift) + g * 4);
    float* p = z + (dst << fshift) + g * 4;
    atomicAdd(p + 0, v.x);
    atomicAdd(p + 1, v.y);
    atomicAdd(p + 2, v.z);
    atomicAdd(p + 3, v.w);
  }
}

// ---------------------------------------------------------------------------
// FP32 WMMA GEMM:  out[N,M] = act(A[N,K] @ W[K,M] + bias),  M == 256.
// WT is W transposed, [M,K] row-major: a lane's B fragment (W rows k,k+1 at
// one column) is one contiguous b64 load. Uniform-base + u32 lane offsets.
// Block = 128 threads (4 waves); block b owns rows [16b,16b+16); wave w owns
// 64 columns as 4 accumulator tiles.
// Modulo software pipeline, period 4: buffer sets S0..S3 hold fragments for
// 4 consecutive k-steps; each sub-step issues its 4 WMMAs and only then
// reloads its own set for k+16, so reloads sit after their consumer (no
// rotation copies) and data is in flight a full period before use.
// N % 16 == 0 (64, 50000 ok). K % 16 == 0, K >= 32 (128, 256 ok).
// ---------------------------------------------------------------------------
__global__ void gemm_bias_act_wmma(const float* __restrict__ A,
                                   const float* __restrict__ WT,
                                   const float* __restrict__ bias,
                                   float* __restrict__ out,
                                   int K, int M, int relu) {
  const int r0      = blockIdx.x * 16;
  const int lane    = threadIdx.x & 31;
  const int half    = lane >> 4;     // 0: K pair {0,1}, 1: K pair {2,3}
  const int lidx    = lane & 15;
  const int colbase = (threadIdx.x >> 5) * 64;

  v8f acc[4] = {};                   // 4 column tiles of 16x16 f32

  const unsigned aoff = (unsigned)(r0 + lidx) * K + 2 * half;
  unsigned boff[4];
#pragma unroll
  for (int t = 0; t < 4; ++t)
    boff[t] = (unsigned)(colbase + t * 16 + lidx) * K + 2 * half;

  // Four pipeline buffer sets: S_i holds fragments for k-step (k + 4*i).
  v2f a[4];
  v2f b[4][4];
#pragma unroll
  for (int i = 0; i < 4; ++i) {
    a[i] = *(const v2f*)(A + aoff + 4 * i);
#pragma unroll
    for (int t = 0; t < 4; ++t) b[i][t] = *(const v2f*)(WT + boff[t] + 4 * i);
  }

  int k = 0;
  for (; k + 32 <= K; k += 16) {
#pragma unroll
    for (int i = 0; i < 4; ++i) {
      // Compute k-step (k + 4*i) with set S_i ...
#pragma unroll
      for (int t = 0; t < 4; ++t)
        acc[t] = __builtin_amdgcn_wmma_f32_16x16x4_f32(
            false, a[i], false, b[i][t], (short)0, acc[t], false, false);
      // ... then refill S_i for k-step (k + 16 + 4*i).
      a[i] = *(const v2f*)(A + aoff + k + 16 + 4 * i);
#pragma unroll
      for (int t = 0; t < 4; ++t)
        b[i][t] = *(const v2f*)(WT + boff[t] + k + 16 + 4 * i);
    }
  }

  // Drain: last 4 k-steps (K-16 .. K-4).
#pragma unroll
  for (int i = 0; i < 4; ++i) {
#pragma unroll
    for (int t = 0; t < 4; ++t)
      acc[t] = __builtin_amdgcn_wmma_f32_16x16x4_f32(
          false, a[i], false, b[i][t], (short)0, acc[t], false, false);
  }

  // Store: C/D layout -> VGPR j holds row M = j + 8*half, col N = lidx.
#pragma unroll
  for (int t = 0; t < 4; ++t) {
    const int col = colbase + t * 16 + lidx;
    const float bv = bias[col];
#pragma unroll
    for (int j = 0; j < 8; ++j) {
      float v = acc[t][j] + bv;
      if (relu) v = fmaxf(v, 0.0f);
      out[(unsigned)(r0 + j + 8 * half) * M + col] = v;
    }
  }
}

// ---------------------------------------------------------------------------
// Column sum (pooling): out[c] += sum over a strip of rows of X[r, c], M=256.
// out must be pre-zeroed. blockDim.x == 256 (one thread per column).
// ---------------------------------------------------------------------------
__global__ void colsum_256(const float* __restrict__ X, int N,
                           float* __restrict__ out) {
  const int ROWS = 128;
  const int col = threadIdx.x;
  const int r0 = blockIdx.x * ROWS;
  int rend = r0 + ROWS;
  if (rend > N) rend = N;
  float s = 0.0f;
  for (int r = r0; r < rend; ++r) s += X[((unsigned)r << 8) + col];
  atomicAdd(&out[col], s);
}

// ---------------------------------------------------------------------------
// Head MLP: h = [qp, dp] (512) -> 256 -> 128 -> 8, ReLU after each layer.
// Single block of 256 threads.
// ---------------------------------------------------------------------------
__global__ void head_mlp(const float* __restrict__ pooled,
                         const float* __restrict__ lW1, const float* __restrict__ lb1,
                         const float* __restrict__ lW2, const float* __restrict__ lb2,
                         const float* __restrict__ lW3, const float* __restrict__ lb3,
                         float* __restrict__ out) {
  __shared__ float h[512];
  __shared__ float h1[256];
  __shared__ float h2[128];
  const int t = threadIdx.x;  // 256 threads

  h[t]       = pooled[t];
  h[t + 256] = pooled[t + 256];
  __syncthreads();

  // Layer 1: 512 -> 256
  {
    float s = lb1[t];
    for (int i = 0; i < 512; ++i) s += h[i] * lW1[i * 256 + t];
    h1[t] = fmaxf(s, 0.0f);
  }
  __syncthreads();

  // Layer 2: 256 -> 128
  if (t < 128) {
    float s = lb2[t];
    for (int i = 0; i < 256; ++i) s += h1[i] * lW2[i * 128 + t];
    h2[t] = fmaxf(s, 0.0f);
  }
  __syncthreads();

  // Layer 3: 128 -> 8
  if (t < 8) {
    float s = lb3[t];
    for (int i = 0; i < 128; ++i) s += h2[i] * lW3[i * 8 + t];
    out[t] = fmaxf(s, 0.0f);
  }
}

// ---------------------------------------------------------------------------
// Host-side orchestration
// ---------------------------------------------------------------------------
static inline int clampGrid(long work, int threads, int maxBlocks) {
  long b = (work + threads - 1) / threads;
  if (b > maxBlocks) b = maxBlocks;
  if (b < 1) b = 1;
  return (int)b;
}

// One full GIN (two GINConv layers) on a graph with N nodes (N % 16 == 0).
// P = W1,b1,W2,b2,W3,b3,W4,b4 (row-major weights; biases used directly);
// WT = transposed weights [M,K] for W1..W4.
static void run_gin(const float* x, const int* edges, int N, int E,
                    const float* const* P, float* const* WT,
                    float* B0, float* B1, float* B2, float* pool_out,
                    hipStream_t stream) {
  const int F = 128, H = 256;

  // ---- conv1: z = x + agg(x) ----
  copy_f4<<<clampGrid((long)N * F / 4, 256, 4096), 256, 0, stream>>>(x, B0, N * F / 4);
  scatter_add_edges<<<clampGrid((long)E * (F / 4), 256, 8192), 256, 0, stream>>>(
      x, edges, E, /*gshift=*/5, B0);
  // a1 = relu(z @ W1 + b1)
  gemm_bias_act_wmma<<<N / 16, 128, 0, stream>>>(B0, WT[0], P[1], B1, F, H, 1);
  // h = relu(a1 @ W2 + b2)   (mlp output + inter-conv ReLU fused)
  gemm_bias_act_wmma<<<N / 16, 128, 0, stream>>>(B1, WT[1], P[3], B2, H, H, 1);

  // ---- conv2: z = h + agg(h) ----
  copy_f4<<<clampGrid((long)N * H / 4, 256, 4096), 256, 0, stream>>>(B2, B0, N * H / 4);
  scatter_add_edges<<<clampGrid((long)E * (H / 4), 256, 8192), 256, 0, stream>>>(
      B2, edges, E, /*gshift=*/6, B0);
  // a2 = relu(z @ W3 + b3)
  gemm_bias_act_wmma<<<N / 16, 128, 0, stream>>>(B0, WT[2], P[5], B1, H, H, 1);
  // out = a2 @ W4 + b4       (no ReLU)
  gemm_bias_act_wmma<<<N / 16, 128, 0, stream>>>(B1, WT[3], P[7], B2, H, H, 0);

  // ---- global add pool ----
  colsum_256<<<(N + 127) / 128, 256, 0, stream>>>(B2, N, pool_out);
}

extern "C" void kernel_launch(void* const* d_in, const int* in_sizes, int n_in,
                              void* d_out, int out_size, void* d_ws, size_t ws_size,
                              hipStream_t stream) {
  (void)in_sizes; (void)n_in; (void)out_size; (void)ws_size;

  const int NQ = 64, EQ = 512;
  const int ND = 50000, ED = 600000;
  const int F = 128, H = 256;

  const float* qx = (const float*)d_in[0];
  const float* dx = (const float*)d_in[1];
  const int*   qe = (const int*)d_in[2];
  const int*   de = (const int*)d_in[3];

  const float* P[22];
  for (int i = 0; i < 22; ++i) P[i] = (const float*)d_in[4 + i];
  // P[0..7]  = qW1,qb1,qW2,qb2,qW3,qb3,qW4,qb4
  // P[8..15] = dW1,db1,dW2,db2,dW3,db3,dW4,db4
  // P[16..21]= lW1,lb1,lW2,lb2,lW3,lb3

  // ---- workspace layout (floats) ----
  float* ws = (float*)d_ws;
  size_t off = 0;
  float* Bq0 = ws + off; off += (size_t)NQ * H;
  float* Bq1 = ws + off; off += (size_t)NQ * H;
  float* Bq2 = ws + off; off += (size_t)NQ * H;
  float* pooled = ws + off; off += 512;              // [qp(256), dp(256)]
  float* WTq[4], *WTd[4];
  WTq[0] = ws + off; off += (size_t)F * H;           // W1^T : [256,128]
  WTq[1] = ws + off; off += (size_t)H * H;
  WTq[2] = ws + off; off += (size_t)H * H;
  WTq[3] = ws + off; off += (size_t)H * H;
  WTd[0] = ws + off; off += (size_t)F * H;
  WTd[1] = ws + off; off += (size_t)H * H;
  WTd[2] = ws + off; off += (size_t)H * H;
  WTd[3] = ws + off; off += (size_t)H * H;
  float* Bd0 = ws + off; off += (size_t)ND * H;
  float* Bd1 = ws + off; off += (size_t)ND * H;
  float* Bd2 = ws + off; off += (size_t)ND * H;

  // ---- weight transposes (tiny, once per call) ----
  {
    dim3 blk(256);
    dim3 g1(F / 16, H / 16);  // W1: [128,256]
    dim3 g2(H / 16, H / 16);  // W2..W4: [256,256]
    transpose_w<<<g1, blk, 0, stream>>>(P[0],  WTq[0], F, H);
    transpose_w<<<g2, blk, 0, stream>>>(P[2],  WTq[1], H, H);
    transpose_w<<<g2, blk, 0, stream>>>(P[4],  WTq[2], H, H);
    transpose_w<<<g2, blk, 0, stream>>>(P[6],  WTq[3], H, H);
    transpose_w<<<g1, blk, 0, stream>>>(P[8],  WTd[0], F, H);
    transpose_w<<<g2, blk, 0, stream>>>(P[10], WTd[1], H, H);
    transpose_w<<<g2, blk, 0, stream>>>(P[12], WTd[2], H, H);
    transpose_w<<<g2, blk, 0, stream>>>(P[14], WTd[3], H, H);
  }

  zero_f<<<1, 512, 0, stream>>>(pooled, 512);

  run_gin(qx, qe, NQ, EQ, &P[0], WTq, Bq0, Bq1, Bq2, pooled, stream);
  run_gin(dx, de, ND, ED, &P[8], WTd, Bd0, Bd1, Bd2, pooled + 256, stream);

  head_mlp<<<1, 256, 0, stream>>>(pooled, P[16], P[17], P[18], P[19], P[20], P[21],
                                  (float*)d_out);
}